// Model_69776038691092
// MI455X (gfx1250) — compile-verified
//
#include <hip/hip_runtime.h>
#include <stdint.h>

// ---------------------------------------------------------------------------
// Bilateral 5x5 filter, 16x3x512x512 f32, reflect padding.
// - Trans-pipe bound (v_exp_f32, 25 exps/pixel). Spatial weight folded into
//   the exp2 argument -> 1 trans + ~5 VALU per tap.
// - Row-pair math packed as 2-wide vectors to pull VALU below the trans
//   floor (targets v_pk_fma_f32 / VOPD dual-issue; TRANS co-executes).
// - Tile staged to LDS via gfx1250 GLOBAL_LOAD_ASYNC_TO_LDS_B32 (ASYNCcnt),
//   window kept in 40 VGPRs so all 100 taps/thread run out of registers.
// ---------------------------------------------------------------------------

#define IMG_H 512
#define IMG_W 512
#define TILE_W 64
#define TILE_H 16
#define HALO 2
#define LW (TILE_W + 2 * HALO) // 68
#define LH (TILE_H + 2 * HALO) // 20
#define NTHREADS 256

typedef float v2f __attribute__((ext_vector_type(2)));

#if defined(__has_builtin)
#if __has_builtin(__builtin_amdgcn_global_load_async_to_lds_b32)
#define USE_ASYNC_LDS 1
#endif
#endif
#ifndef USE_ASYNC_LDS
#define USE_ASYNC_LDS 0
#endif

__device__ __forceinline__ void copy_g2lds(const float* __restrict__ g, float* l) {
#if USE_ASYNC_LDS
  // gfx1250: GLOBAL_LOAD_ASYNC_TO_LDS_B32 (per-lane global -> LDS, ASYNCcnt)
  __builtin_amdgcn_global_load_async_to_lds_b32(
      (__attribute__((address_space(1))) int*)(uintptr_t)g,
      (__attribute__((address_space(3))) int*)(uint32_t)(uintptr_t)l,
      /*offset=*/0, /*cpol=*/0);
#else
  *l = *g;
#endif
}

__device__ __forceinline__ void wait_async_lds() {
#if USE_ASYNC_LDS
#if __has_builtin(__builtin_amdgcn_s_wait_asynccnt)
  __builtin_amdgcn_s_wait_asynccnt(0);
#else
  asm volatile("s_wait_asynccnt 0" ::: "memory");
#endif
#endif
}

__device__ __forceinline__ int reflect_idx(int i, int n) {
  // jnp.pad mode="reflect": -1 -> 1, -2 -> 2, n -> n-2, n+1 -> n-3
  i = (i < 0) ? -i : i;
  i = (i >= n) ? (2 * n - 2 - i) : i;
  return i;
}

__global__ __launch_bounds__(NTHREADS) void bilateral5x5_kernel(
    const float* __restrict__ img, float* __restrict__ out) {
  __shared__ float tile[LH][LW];

  const int plane = blockIdx.z;                  // b*C + c, 0..47
  const int tx0 = blockIdx.x * TILE_W;           // tile origin x
  const int ty0 = blockIdx.y * TILE_H;           // tile origin y
  const size_t plane_off = (size_t)plane * (IMG_H * IMG_W);
  const float* __restrict__ src = img + plane_off;
  const int tid = threadIdx.x;

  // ---- Stage (TILE+halo) tile into LDS with reflect indexing (async) ----
  float* lds_flat = &tile[0][0];
  for (int idx = tid; idx < LH * LW; idx += NTHREADS) {
    const int ly = idx / LW;
    const int lx = idx - ly * LW;
    const int gy = reflect_idx(ty0 + ly - HALO, IMG_H);
    const int gx = reflect_idx(tx0 + lx - HALO, IMG_W);
    copy_g2lds(&src[gy * IMG_W + gx], &lds_flat[idx]);
  }
  wait_async_lds();
  __syncthreads();

  // ---- Each thread: 1x4 vertical strip of outputs from a 8x5 reg window ----
  const int tx = tid & (TILE_W - 1); // 0..63 (consecutive within a wave32)
  const int tg = tid >> 6;           // 0..3
  const int j0 = tg * 4;             // first tile-row of this thread's strip

  float win[8][5];
#pragma unroll
  for (int r = 0; r < 8; ++r)
#pragma unroll
    for (int c = 0; c < 5; ++c)
      win[r][c] = tile[j0 + r][tx + c];

  // exp2 folding: w = spatial * exp(-diff^2 * 50)
  //             = exp2( -d2*(log2e/8)  -  diff^2 * 50*log2e )
  const float kRange = 72.13475204444817f; // 50 * log2(e)

  float res[4];
#pragma unroll
  for (int p = 0; p < 2; ++p) { // two packed row-pairs: rows (2p, 2p+1)
    const int r0 = 2 * p;
    v2f cen;
    cen.x = win[r0 + 2][2];
    cen.y = win[r0 + 3][2];
    v2f acc = (v2f)(0.0f);
    v2f nrm = (v2f)(0.0f);
#pragma unroll
    for (int dy = 0; dy < 5; ++dy) {
#pragma unroll
      for (int dx = 0; dx < 5; ++dx) {
        const int d2 = (dy - 2) * (dy - 2) + (dx - 2) * (dx - 2);
        const float c2 = -(float)d2 * 0.18033688011112043f; // log2(e)/8
        v2f nb;
        nb.x = win[r0 + dy][dx];
        nb.y = win[r0 + 1 + dy][dx];
        const v2f diff = cen - nb;
        const v2f t = diff * diff;
        const v2f a =
            __builtin_elementwise_fma(t, (v2f)(-kRange), (v2f)(c2));
        v2f e;
        e.x = __builtin_amdgcn_exp2f(a.x);
        e.y = __builtin_amdgcn_exp2f(a.y);
        acc = __builtin_elementwise_fma(e, nb, acc);
        nrm += e;
      }
    }
    res[r0 + 0] = acc.x * __builtin_amdgcn_rcpf(nrm.x + 1e-8f);
    res[r0 + 1] = acc.y * __builtin_amdgcn_rcpf(nrm.y + 1e-8f);
  }

  float* __restrict__ dst = out + plane_off;
  const int ox = tx0 + tx;
#pragma unroll
  for (int r = 0; r < 4; ++r)
    dst[(size_t)(ty0 + j0 + r) * IMG_W + ox] = res[r];
}

extern "C" void kernel_launch(void* const* d_in, const int* in_sizes, int n_in,
                              void* d_out, int out_size, void* d_ws, size_t ws_size,
                              hipStream_t stream) {
  (void)n_in;
  (void)d_ws;
  (void)ws_size;
  (void)out_size;
  const float* img = (const float*)d_in[0];
  float* out = (float*)d_out;
  const int planes = in_sizes[0] / (IMG_H * IMG_W); // 16*3 = 48
  dim3 grid(IMG_W / TILE_W, IMG_H / TILE_H, planes);
  dim3 block(NTHREADS, 1, 1);
  bilateral5x5_kernel<<<grid, block, 0, stream>>>(img, out);
}